// GRUKerasModel_12378095747944
// MI455X (gfx1250) — compile-verified
//
#include <hip/hip_runtime.h>
#include <hip/hip_bf16.h>
#include <cstddef>
#include <cstdint>

typedef __attribute__((ext_vector_type(2))) float v2f;
typedef __attribute__((ext_vector_type(8))) float v8f;

#define GB 256      // batch
#define GT 2048     // timesteps
#define GF 64       // features
#define GU 64       // units
#define G3U 192     // 3*U
#define HS 66       // LDS row stride (even, odd in b64 units -> conflict-free)

__device__ __forceinline__ float sig_f(float x) {
    return 1.0f / (1.0f + __expf(-x));
}
__device__ __forceinline__ float tanh_f(float x) {
    float e = __expf(2.0f * x);     // saturates correctly at +/-inf
    return 1.0f - 2.0f / (e + 1.0f);
}

// CDNA5 async copy: global -> LDS, 16 bytes per lane, tracked by ASYNCcnt
__device__ __forceinline__ void async_ld16(uint32_t lds_addr, const void* g) {
    asm volatile("global_load_async_to_lds_b128 %0, %1, off"
                 :: "v"(lds_addr), "v"(g) : "memory");
}
__device__ __forceinline__ void wait_async0() {
    asm volatile("s_wait_asynccnt 0x0" ::: "memory");
}

// ---------------------------------------------------------------------------
// Phase 1: x_proj = inputs @ kernel + bias[0], written TRANSPOSED as
// xw[t][b][0:192] so the scan consumes one contiguous 12 KB block per step.
// Flattened 1D grid with the 12 n-tiles CONSECUTIVE for each m-group, so the
// 16 KB A-block is L0/L2-hot across all 12 n passes (A traffic: 1x not 12x).
// ---------------------------------------------------------------------------
__global__ void __launch_bounds__(128)
xproj_kernel(const float* __restrict__ inp,    // [BT, 64]
             const float* __restrict__ wk,     // [64, 192]
             const float* __restrict__ bias,   // [2, 192] (row 0)
             float* __restrict__ xw)           // [T, B, 192]
{
    const int lane = threadIdx.x & 31;
    const int wave = threadIdx.x >> 5;
    const int nt      = (int)blockIdx.x % 12;       // n-tile: fastest
    const int m_group = (int)blockIdx.x / 12;       // group of 4 m-tiles
    const int m0 = (m_group * 4 + wave) * 16;
    const int n0 = nt * 16;
    const int mrow = lane & 15;
    const int kh   = lane >> 4;
    const int ncol = lane & 15;

    const float* __restrict__ arow = inp + (size_t)(m0 + mrow) * GF;

    v8f c = {};
#pragma unroll
    for (int kb = 0; kb < GF; kb += 4) {
        const int k0 = kb + 2 * kh;
        v2f a, b;
        a.x = arow[k0];
        a.y = arow[k0 + 1];
        b.x = wk[(size_t)k0 * G3U + n0 + ncol];
        b.y = wk[(size_t)(k0 + 1) * G3U + n0 + ncol];
        c = __builtin_amdgcn_wmma_f32_16x16x4_f32(
                false, a, false, b, (short)0, c, false, false);
    }

    const float bs = bias[n0 + ncol];
#pragma unroll
    for (int v = 0; v < 8; ++v) {
        const int m = m0 + v + 8 * kh;       // m = b*T + t
        const int b_ = m >> 11;              // / GT
        const int tt = m & (GT - 1);         // % GT
        xw[((size_t)tt * GB + b_) * G3U + n0 + ncol] = c[v] + bs;
    }
}

// ---------------------------------------------------------------------------
// Phase 2: GRU scan. 16 independent workgroups x 128 threads (4 waves);
// workgroup owns 16 batch rows. Recurrent weight fragments + h-state + masks
// live in REGISTERS; pre-masked h lives in double-buffered LDS; x_proj is
// async-copied into LDS one timestep ahead. One barrier per step.
// ---------------------------------------------------------------------------
__global__ void __launch_bounds__(128)
gru_scan_kernel(const float* __restrict__ xw,      // [T, B, 192]
                const float* __restrict__ rk,      // [64, 192]
                const float* __restrict__ bias,    // [2, 192] (row 1)
                const float* __restrict__ rmask,   // [3, B, 64]
                const float* __restrict__ dw,      // [64]
                const float* __restrict__ db,      // [1]
                float* __restrict__ out)           // [B]
{
    __shared__ __align__(16) float hz[2][16][HS];  // h*mz (double buffered)
    __shared__ __align__(16) float hr[2][16][HS];  // h*mr
    __shared__ __align__(16) float hm[2][16][HS];  // h*mh
    __shared__ __align__(16) float xs[2][16 * G3U]; // x_proj stage, 2x12 KB

    const int tid = threadIdx.x;
    const int blk = blockIdx.x;
    const int brow0 = blk * 16;
    const int lane = tid & 31;
    const int j    = tid >> 5;          // wave id = N-tile
    const int n0   = j * 16;
    const int mrow = lane & 15;
    const int kh   = lane >> 4;
    const int col  = n0 + (lane & 15);

    // h0 = 0 -> masked copies are 0
    for (int idx = tid; idx < 16 * HS; idx += 128) {
        hz[0][0][idx] = 0.0f; hr[0][0][idx] = 0.0f; hm[0][0][idx] = 0.0f;
    }

    // ---- recurrent-weight B fragments: registers for the whole scan ----
    v2f wz[16], wr[16], wh[16];
#pragma unroll
    for (int kk = 0; kk < 16; ++kk) {
        const int k0 = 4 * kk + 2 * kh;
        wz[kk].x = rk[(size_t)k0 * G3U + 0 * GU + col];
        wz[kk].y = rk[(size_t)(k0 + 1) * G3U + 0 * GU + col];
        wr[kk].x = rk[(size_t)k0 * G3U + 1 * GU + col];
        wr[kk].y = rk[(size_t)(k0 + 1) * G3U + 1 * GU + col];
        wh[kk].x = rk[(size_t)k0 * G3U + 2 * GU + col];
        wh[kk].y = rk[(size_t)(k0 + 1) * G3U + 2 * GU + col];
    }
    const float bz  = bias[G3U + 0 * GU + col];
    const float brr = bias[G3U + 1 * GU + col];
    const float bh  = bias[G3U + 2 * GU + col];

    // ---- per-lane h state + dropout masks in registers ----
    float h_reg[8], mz[8], mr[8], mh[8];
#pragma unroll
    for (int v = 0; v < 8; ++v) {
        const int row = v + 8 * kh;
        const size_t mi = (size_t)(brow0 + row) * GU + col;
        mz[v] = rmask[0 * GB * GU + mi];
        mr[v] = rmask[1 * GB * GU + mi];
        mh[v] = rmask[2 * GB * GU + mi];
        h_reg[v] = 0.0f;
    }

    // ---- async preload x_proj for t=0 ----
    const uint32_t lds0 = (uint32_t)(uintptr_t)(void*)&xs[0][0];
    const uint32_t lds1 = (uint32_t)(uintptr_t)(void*)&xs[1][0];
    {
        const char* g = (const char*)(xw + (size_t)brow0 * G3U);
#pragma unroll
        for (int cch = 0; cch < 6; ++cch) {
            const uint32_t off = (uint32_t)tid * 16u + (uint32_t)cch * 2048u;
            async_ld16(lds0 + off, g + off);
        }
    }
    wait_async0();
    __syncthreads();

    for (int t = 0; t < GT; ++t) {
        // kick off async copy of x_proj for t+1
        if (t + 1 < GT) {
            const char* g =
                (const char*)(xw + ((size_t)(t + 1) * GB + brow0) * G3U);
            const uint32_t l = ((t + 1) & 1) ? lds1 : lds0;
#pragma unroll
            for (int cch = 0; cch < 6; ++cch) {
                const uint32_t off = (uint32_t)tid * 16u + (uint32_t)cch * 2048u;
                async_ld16(l + off, g + off);
            }
        }

        const float (*__restrict__ hzL)[HS] = hz[t & 1];
        const float (*__restrict__ hrL)[HS] = hr[t & 1];
        const float (*__restrict__ hmL)[HS] = hm[t & 1];
        float (*__restrict__ hzN)[HS] = hz[(t + 1) & 1];
        float (*__restrict__ hrN)[HS] = hr[(t + 1) & 1];
        float (*__restrict__ hmN)[HS] = hm[(t + 1) & 1];

        // 6 independent WMMA chains (2 per gate, split over K)
        v8f cz0 = {}, cz1 = {}, cr0 = {}, cr1 = {}, ch0 = {}, ch1 = {};
#pragma unroll
        for (int kk = 0; kk < 8; ++kk) {
            const int k0 = 4 * kk + 2 * kh;
            const v2f az = *(const v2f*)&hzL[mrow][k0];
            const v2f ar = *(const v2f*)&hrL[mrow][k0];
            const v2f ah = *(const v2f*)&hmL[mrow][k0];
            cz0 = __builtin_amdgcn_wmma_f32_16x16x4_f32(
                      false, az, false, wz[kk], (short)0, cz0, false, false);
            cr0 = __builtin_amdgcn_wmma_f32_16x16x4_f32(
                      false, ar, false, wr[kk], (short)0, cr0, false, false);
            ch0 = __builtin_amdgcn_wmma_f32_16x16x4_f32(
                      false, ah, false, wh[kk], (short)0, ch0, false, false);
        }
#pragma unroll
        for (int kk = 8; kk < 16; ++kk) {
            const int k0 = 4 * kk + 2 * kh;
            const v2f az = *(const v2f*)&hzL[mrow][k0];
            const v2f ar = *(const v2f*)&hrL[mrow][k0];
            const v2f ah = *(const v2f*)&hmL[mrow][k0];
            cz1 = __builtin_amdgcn_wmma_f32_16x16x4_f32(
                      false, az, false, wz[kk], (short)0, cz1, false, false);
            cr1 = __builtin_amdgcn_wmma_f32_16x16x4_f32(
                      false, ar, false, wr[kk], (short)0, cr1, false, false);
            ch1 = __builtin_amdgcn_wmma_f32_16x16x4_f32(
                      false, ah, false, wh[kk], (short)0, ch1, false, false);
        }

        // ---- gates + state update; write pre-masked h for next step ----
        const float* __restrict__ xsL = xs[t & 1];
#pragma unroll
        for (int v = 0; v < 8; ++v) {
            const int row = v + 8 * kh;
            const float xz = xsL[row * G3U + 0 * GU + col];
            const float xr = xsL[row * G3U + 1 * GU + col];
            const float xh = xsL[row * G3U + 2 * GU + col];
            const float z  = sig_f(cz0[v] + cz1[v] + xz + bz);
            const float r  = sig_f(cr0[v] + cr1[v] + xr + brr);
            const float hhat = tanh_f(xh + r * (ch0[v] + ch1[v] + bh));
            const float hnew = z * h_reg[v] + (1.0f - z) * hhat;
            h_reg[v] = hnew;
            hzN[row][col] = hnew * mz[v];
            hrN[row][col] = hnew * mr[v];
            hmN[row][col] = hnew * mh[v];
        }

        wait_async0();     // t+1 stage complete before the barrier publishes it
        __syncthreads();
    }

    // ---- fused final dense: out[b] = h_last[b,:] . dw + db ----
#pragma unroll
    for (int v = 0; v < 8; ++v) hz[0][v + 8 * kh][col] = h_reg[v];
    __syncthreads();
    if (tid < 16) {
        float acc = db[0];
#pragma unroll 8
        for (int u = 0; u < GU; ++u) acc += hz[0][tid][u] * dw[u];
        out[brow0 + tid] = acc;
    }
}

// ---------------------------------------------------------------------------
extern "C" void kernel_launch(void* const* d_in, const int* in_sizes, int n_in,
                              void* d_out, int out_size, void* d_ws, size_t ws_size,
                              hipStream_t stream) {
    const float* inputs  = (const float*)d_in[0];   // [B,T,F]
    const float* kernelW = (const float*)d_in[1];   // [F,3U]
    const float* rk      = (const float*)d_in[2];   // [U,3U]
    const float* bias    = (const float*)d_in[3];   // [2,3U]
    const float* dw      = (const float*)d_in[4];   // [U,1]
    const float* db      = (const float*)d_in[5];   // [1]
    const float* rmask   = (const float*)d_in[6];   // [3,B,U]
    float* out = (float*)d_out;                     // [B,1]
    float* xw  = (float*)d_ws;                      // [T, B, 192] scratch

    // 1D grid: 12 n-tiles consecutive per m-group -> A-block reuse in L0/L2
    const int nblocks = ((GB * GT) / 64) * 12;      // 8192 m-groups * 12
    xproj_kernel<<<nblocks, 128, 0, stream>>>(inputs, kernelW, bias, xw);
    gru_scan_kernel<<<16, 128, 0, stream>>>(xw, rk, bias, rmask, dw, db, out);
}